// Detector_43413529428202
// MI455X (gfx1250) — compile-verified
//
#include <hip/hip_runtime.h>
#include <hip/hip_bf16.h>
#include <cstdint>

// ---------------------------------------------------------------------------
// YOLO-style detection decode for MI455X (gfx1250).
//
// Memory-bound: ~116 MB read + ~8 MB write => ~5.3 us at 23.3 TB/s HBM.
// No GEMM structure => no WMMA. The CDNA5 feature that fits is the Tensor
// Data Mover: each block's input is an [85 channels x 128 positions] tile of
// the [C, H*W] plane with row stride H*W -- a textbook 2D strided TDM tile.
// Wave 0 of each block issues one TENSOR_LOAD_TO_LDS, waits on TENSORcnt,
// then all 128 threads decode from LDS (bank-conflict-free columns).
//
// Grid is (tiles, 3 anchors, B) and W/NHW are template constants so block
// decomposition and p/W, p%W cost no runtime integer division.
// ---------------------------------------------------------------------------

typedef __attribute__((ext_vector_type(4))) unsigned int v4u;
typedef __attribute__((ext_vector_type(8))) int          v8i;
typedef __attribute__((ext_vector_type(4))) int          v4i;

#define TILE 128     // spatial positions per block
#define NCH   85     // channels per anchor (5 + 80 classes)
#define NCLS  80

template <int W_, int NHW_>
__global__ __launch_bounds__(TILE)
void decode_scale_kernel(const float* __restrict__ in,       // [B,255,H,W]
                         const float* __restrict__ anchors,  // this scale: [3][2]
                         float* __restrict__ out,            // full output base
                         float step, float thresh,
                         long long outBaseRow)
{
    __shared__ float sbuf[NCH * TILE];   // 85*128*4 = 43.5 KB

    const int tile = blockIdx.x;
    const int a    = blockIdx.y;
    const int b    = blockIdx.z;

    const int p0        = tile * TILE;
    const int remaining = NHW_ - p0;     // > 0 by construction

    // ---- wave 0: one TDM descriptor fetching [NCH x TILE] into LDS --------
    if ((threadIdx.x >> 5) == 0) {
        const uint64_t ga =
            (uint64_t)(uintptr_t)(in + ((size_t)b * 255 + (size_t)a * NCH) * (size_t)NHW_ + p0);
        const uint32_t lds = (uint32_t)(uintptr_t)(void*)sbuf;  // generic->LDS offset (low 32b)

        // D# group 0 (128b): count=1 | lds_addr | global_addr[56:0] | type=2
        v4u g0;
        g0[0] = 1u;
        g0[1] = lds;
        g0[2] = (uint32_t)(ga & 0xFFFFFFFFu);
        g0[3] = (uint32_t)((ga >> 32) & 0x01FFFFFFu) | 0x80000000u;

        // D# group 1 (256b)
        const uint32_t td0 = (uint32_t)remaining;   // row extent from tile start (OOB -> 0)
        v8i g1;
        g1[0] = (int)(2u << 16);                                  // data_size=4B, no mask/flags
        g1[1] = (int)((td0 & 0xFFFFu) << 16);                     // tensor_dim0[15:0]
        g1[2] = (int)(((td0 >> 16) & 0xFFFFu) | ((uint32_t)NCH << 16)); // tensor_dim0[31:16] | tensor_dim1=85
        g1[3] = (int)((uint32_t)TILE << 16);                      // tile_dim0=TILE
        g1[4] = (int)(uint32_t)NCH;                               // tile_dim1=85, tile_dim2=0
        g1[5] = (int)(uint32_t)NHW_;                              // tensor_dim0_stride
        g1[6] = 0;
        g1[7] = 0;

        const v4i gz4 = {0, 0, 0, 0};                  // groups 2/3 unused (2D tile)
        const v8i gz8 = {0, 0, 0, 0, 0, 0, 0, 0};

        __builtin_amdgcn_tensor_load_to_lds(g0, g1, gz4, gz4, gz8, 0);
        __builtin_amdgcn_s_wait_tensorcnt(0);
    }
    __syncthreads();

    // ---- per-thread decode of one (b, p, a) cell ---------------------------
    const int t = threadIdx.x;
    const int p = p0 + t;
    if (p >= NHW_) return;

    const long long row = outBaseRow + ((long long)b * NHW_ + p) * 3 + a;
    float* o = out + row * 6;

    const float conf = sbuf[0 * TILE + t];
    if (!(conf > thresh)) {
        // masked row -> zeros (d_out is poisoned; must write explicitly)
        const float2 z = make_float2(0.f, 0.f);
        *(float2*)(o + 0) = z;
        *(float2*)(o + 2) = z;
        *(float2*)(o + 4) = z;
        return;
    }

    const float tx = sbuf[1 * TILE + t];
    const float ty = sbuf[2 * TILE + t];
    const float tw = sbuf[3 * TILE + t];
    const float th = sbuf[4 * TILE + t];

    const float gy = (float)(p / W_);    // constant divisor -> mul/shift
    const float gx = (float)(p % W_);
    const float aw = anchors[a * 2 + 0];
    const float ah = anchors[a * 2 + 1];

    const float cx = (gx + tx) * step;
    const float cy = (gy + ty) * step;
    const float w  = aw * __expf(tw);
    const float h  = ah * __expf(th);
    const float x1 = cx - w * 0.5f;
    const float y1 = cy - h * 0.5f;

    // argmax over 80 classes; first-max-wins matches jnp.argmax tie rule
    float best = sbuf[5 * TILE + t];
    int   bi   = 0;
#pragma unroll 8
    for (int k = 1; k < NCLS; ++k) {
        const float v = sbuf[(5 + k) * TILE + t];
        if (v > best) { best = v; bi = k; }
    }

    o[0] = conf;
    o[1] = x1;
    o[2] = y1;
    o[3] = x1 + w;
    o[4] = y1 + h;
    o[5] = (float)bi;
}

extern "C" void kernel_launch(void* const* d_in, const int* in_sizes, int n_in,
                              void* d_out, int out_size, void* d_ws, size_t ws_size,
                              hipStream_t stream) {
    const float* in13 = (const float*)d_in[0];
    const float* in26 = (const float*)d_in[1];
    const float* in52 = (const float*)d_in[2];
    const float* anc  = (const float*)d_in[3];   // [3][3][2]
    float* out = (float*)d_out;

    const int B = in_sizes[0] / (255 * 13 * 13);

    long long baseRow = 0;

    { // scale 13: step 32, thresh 0.5
        constexpr int NHW = 13 * 13, tiles = (NHW + TILE - 1) / TILE;
        decode_scale_kernel<13, NHW><<<dim3(tiles, 3, B), TILE, 0, stream>>>(
            in13, anc + 0, out, 32.0f, 0.5f, baseRow);
        baseRow += (long long)B * NHW * 3;
    }
    { // scale 26: step 16, thresh 0.5
        constexpr int NHW = 26 * 26, tiles = (NHW + TILE - 1) / TILE;
        decode_scale_kernel<26, NHW><<<dim3(tiles, 3, B), TILE, 0, stream>>>(
            in26, anc + 6, out, 16.0f, 0.5f, baseRow);
        baseRow += (long long)B * NHW * 3;
    }
    { // scale 52: step 8, thresh 0.9
        constexpr int NHW = 52 * 52, tiles = (NHW + TILE - 1) / TILE;
        decode_scale_kernel<52, NHW><<<dim3(tiles, 3, B), TILE, 0, stream>>>(
            in52, anc + 12, out, 8.0f, 0.9f, baseRow);
    }
}